// ApproximateAttention_25443386261620
// MI455X (gfx1250) — compile-verified
//
#include <hip/hip_runtime.h>
#include <hip/hip_bf16.h>
#include <math.h>

#define NHASH 8
#define BKT   64      // BUCKET_SIZE == n_buckets
#define D     64
#define HD    67      // d + 3 (ALSH extension)
#define LDSD  76      // padded LDS row stride in floats (16B aligned, conflict-friendly)

typedef __attribute__((ext_vector_type(2))) float v2f;
typedef __attribute__((ext_vector_type(8))) float v8f;

// ---------------------------------------------------------------------------
// Stage 1: ALSH extension + L2 hash.  grid = 2*bs*(S/64) blocks, 512 threads.
// First half of grid hashes queries (mode 0), second half keys (mode 1).
// Each thread owns one of the 512 projection columns (alpha column cached in
// registers) and iterates over the 64 positions staged in LDS.
// ---------------------------------------------------------------------------
__global__ __launch_bounds__(512)
void alsh_hash_kernel(const float* __restrict__ query,
                      const float* __restrict__ keym,
                      const float* __restrict__ alpha_q, const float* __restrict__ beta_q,
                      const float* __restrict__ w_q,
                      const float* __restrict__ alpha_k, const float* __restrict__ beta_k,
                      const float* __restrict__ w_k,
                      int* __restrict__ qbucket, int* __restrict__ kbucket,
                      int bs, int S)
{
    const int nPerMode = bs * (S >> 6);
    const int blk  = blockIdx.x;
    const int mode = (blk >= nPerMode) ? 1 : 0;
    const int id   = mode ? (blk - nPerMode) : blk;
    const int b    = id / (S >> 6);
    const int pos0 = (id % (S >> 6)) << 6;

    const float* __restrict__ x    = mode ? keym    : query;
    const float* __restrict__ alph = mode ? alpha_k : alpha_q;
    const float* __restrict__ beta = mode ? beta_k  : beta_q;
    const float* __restrict__ w    = mode ? w_k     : w_q;
    int* __restrict__ bucket       = mode ? kbucket : qbucket;

    __shared__ float s_vec[64][HD + 1];
    __shared__ float s_red[16];

    const int t    = threadIdx.x;
    const int lane = t & 31;
    const int wv   = t >> 5;
    const int HW   = NHASH * BKT;   // 512 == blockDim.x

    // cooperative load of the 64x64 tile of x
    for (int i = 0; i < 8; ++i) {
        int idx = t + 512 * i;
        int r = idx >> 6, c = idx & 63;
        s_vec[r][c] = x[((size_t)b * S + pos0 + r) * D + c];
    }
    __syncthreads();

    // per-row ALSH extension
    if (t < 64) {
        float n2 = 0.f;
        for (int c = 0; c < D; ++c) { float v = s_vec[t][c]; n2 += v * v; }
        if (mode == 0) {
            float inv = rsqrtf(n2);
            for (int c = 0; c < D; ++c) s_vec[t][c] *= inv;
            s_vec[t][64] = 0.5f; s_vec[t][65] = 0.5f; s_vec[t][66] = 0.5f;
        } else {
            float n4 = n2 * n2;
            s_vec[t][64] = n2; s_vec[t][65] = n4; s_vec[t][66] = n4 * n4;
        }
    }
    __syncthreads();

    // register-cache this thread's alpha column (67 floats)
    float areg[HD];
    for (int k = 0; k < HD; ++k) areg[k] = alph[k * HW + t];
    const float bj = beta[t];
    const float wj = w[t & (BKT - 1)];

    for (int p = 0; p < 64; ++p) {
        float acc = bj;
        #pragma unroll
        for (int k = 0; k < HD; ++k) acc += areg[k] * s_vec[p][k];
        float part = floorf(acc / 2.5f) * wj;     // R = 2.5
        part += __shfl_xor(part, 1);
        part += __shfl_xor(part, 2);
        part += __shfl_xor(part, 4);
        part += __shfl_xor(part, 8);
        part += __shfl_xor(part, 16);
        if (lane == 0) s_red[wv] = part;
        __syncthreads();
        if (t < NHASH) {
            // hash h groups columns [64h, 64h+64) == waves 2h, 2h+1
            float proj = s_red[2 * t] + s_red[2 * t + 1];
            int ti = (int)proj;                    // trunc toward zero (astype int32)
            int m = ti % BKT; if (m < 0) m += BKT; // python-style mod
            bucket[((size_t)(b * NHASH + t)) * S + pos0 + p] = m;
        }
        __syncthreads();
    }
}

// ---------------------------------------------------------------------------
// Stage 2: stable counting sort of positions by bucket per (b, hash).
// grid = 2*bs*8 blocks (q then k), 256 threads.
// ---------------------------------------------------------------------------
__global__ __launch_bounds__(256)
void stable_bucket_sort_kernel(const int* __restrict__ qbucket,
                               const int* __restrict__ kbucket,
                               int* __restrict__ qsorted, int* __restrict__ ksorted,
                               int bs, int S)
{
    const int nBH = bs * NHASH;
    const int blk = blockIdx.x;
    const int* __restrict__ bucket = (blk < nBH) ? qbucket : kbucket;
    int* __restrict__ sorted       = (blk < nBH) ? qsorted : ksorted;
    const size_t base = (size_t)((blk < nBH) ? blk : blk - nBH) * S;

    __shared__ int s_b[4096];
    __shared__ int s_cnt[BKT];
    __shared__ int s_base[BKT];

    const int t = threadIdx.x;
    if (t < BKT) s_cnt[t] = 0;
    __syncthreads();
    for (int i = t; i < S; i += 256) s_b[i] = bucket[base + i];
    __syncthreads();
    for (int i = t; i < S; i += 256) atomicAdd(&s_cnt[s_b[i]], 1);
    __syncthreads();
    if (t == 0) {
        int run = 0;
        for (int k = 0; k < BKT; ++k) { s_base[k] = run; run += s_cnt[k]; }
    }
    __syncthreads();
    if (t < BKT) {
        int idx = s_base[t];
        for (int p = 0; p < S; ++p)
            if (s_b[p] == t) sorted[base + idx++] = p;   // stable by position
    }
}

// ---------------------------------------------------------------------------
// Stage 3: per-chunk 64x64x64 attention with V_WMMA_F32_16X16X4_F32.
// grid = bs*8*(S/64) blocks, 128 threads (4 waves).  Wave w owns score rows
// [16w,16w+16) as four 16x16 f32 WMMA tiles.
// ---------------------------------------------------------------------------
__global__ __launch_bounds__(128)
void lsh_attn_chunk_kernel(const float* __restrict__ query,
                           const float* __restrict__ keym,
                           const float* __restrict__ value,
                           const int* __restrict__ qsorted,
                           const int* __restrict__ ksorted,
                           float* __restrict__ logits,
                           float* __restrict__ obuf,
                           int bs, int S)
{
    const int nChunks = S >> 6;
    const int blk   = blockIdx.x;
    const int chunk = blk % nChunks;
    const int bh    = blk / nChunks;     // b*8 + h
    const int b     = bh / NHASH;

    __shared__ float sQ[64 * LDSD];      // reused for P after softmax
    __shared__ float sK[64 * LDSD];      // reused for O before scatter
    __shared__ float sV[64 * LDSD];
    __shared__ int   sQI[64];
    __shared__ int   sKI[64];

    const int t    = threadIdx.x;
    const int lane = t & 31;
    const int wv   = t >> 5;

    const size_t bhS = (size_t)bh * S;
    if (t < 64) sQI[t]      = qsorted[bhS + chunk * 64 + t];
    else        sKI[t - 64] = ksorted[bhS + chunk * 64 + (t - 64)];
    __syncthreads();

    // gather sorted Q/K/V rows into LDS (float4, coalesced rows)
    {
        const int c4 = t & 15;
        const int r0 = t >> 4;
        for (int i = 0; i < 8; ++i) {
            int r = r0 + 8 * i;
            size_t qrow = ((size_t)b * S + sQI[r]) * D;
            size_t krow = ((size_t)b * S + sKI[r]) * D;
            float4 qv = ((const float4*)(query + qrow))[c4];
            float4 kv = ((const float4*)(keym  + krow))[c4];
            float4 vv = ((const float4*)(value + krow))[c4];
            *(float4*)&sQ[r * LDSD + c4 * 4] = qv;
            *(float4*)&sK[r * LDSD + c4 * 4] = kv;
            *(float4*)&sV[r * LDSD + c4 * 4] = vv;
        }
    }
    __syncthreads();

    const int col   = lane & 15;
    const int khalf = (lane >> 4) << 1;      // lanes 16-31 hold K+2,K+3
    const int m     = wv * 16 + col;         // A-matrix row for this lane

    // scores = Q @ K^T  (16x64 strip per wave, K accumulated 4 at a time)
    v8f acc[4] = {};
    for (int kb = 0; kb < 16; ++kb) {
        const int kk = kb * 4 + khalf;
        v2f a;
        a.x = sQ[m * LDSD + kk];
        a.y = sQ[m * LDSD + kk + 1];
        #pragma unroll
        for (int ni = 0; ni < 4; ++ni) {
            v2f bm;                                      // B[k][n] = K[n][k]
            bm.x = sK[(ni * 16 + col) * LDSD + kk];
            bm.y = sK[(ni * 16 + col) * LDSD + kk + 1];
            acc[ni] = __builtin_amdgcn_wmma_f32_16x16x4_f32(
                false, a, false, bm, (short)0, acc[ni], false, false);
        }
    }

    // row softmax + logsumexp; write P into sQ (wave-private rows -> no barrier)
    #pragma unroll
    for (int e = 0; e < 8; ++e) {
        float mx = fmaxf(fmaxf(acc[0][e], acc[1][e]), fmaxf(acc[2][e], acc[3][e]));
        mx = fmaxf(mx, __shfl_xor(mx, 1));
        mx = fmaxf(mx, __shfl_xor(mx, 2));
        mx = fmaxf(mx, __shfl_xor(mx, 4));
        mx = fmaxf(mx, __shfl_xor(mx, 8));   // reduce within 16-lane half
        float ex[4], s = 0.f;
        #pragma unroll
        for (int ni = 0; ni < 4; ++ni) { ex[ni] = __expf(acc[ni][e] - mx); s += ex[ni]; }
        s += __shfl_xor(s, 1);
        s += __shfl_xor(s, 2);
        s += __shfl_xor(s, 4);
        s += __shfl_xor(s, 8);
        const float inv = 1.0f / s;
        const int row = wv * 16 + e + ((lane >> 4) << 3);   // C/D tile row layout
        #pragma unroll
        for (int ni = 0; ni < 4; ++ni)
            sQ[row * LDSD + ni * 16 + col] = ex[ni] * inv;
        if (col == 0)
            logits[bhS + sQI[row]] = mx + __logf(s);        // logsumexp, scattered
    }

    // O = P @ V (reads only wave-private P rows + sV -> no barrier needed)
    v8f oacc[4] = {};
    for (int kb = 0; kb < 16; ++kb) {
        const int kk = kb * 4 + khalf;
        v2f a;
        a.x = sQ[m * LDSD + kk];
        a.y = sQ[m * LDSD + kk + 1];
        #pragma unroll
        for (int ni = 0; ni < 4; ++ni) {
            v2f bm;                                      // B[k][n] = V[k][n]
            bm.x = sV[kk * LDSD + ni * 16 + col];
            bm.y = sV[(kk + 1) * LDSD + ni * 16 + col];
            oacc[ni] = __builtin_amdgcn_wmma_f32_16x16x4_f32(
                false, a, false, bm, (short)0, oacc[ni], false, false);
        }
    }

    // stage O in sK, then scatter rows to unsorted positions
    #pragma unroll
    for (int e = 0; e < 8; ++e) {
        const int row = wv * 16 + e + ((lane >> 4) << 3);
        #pragma unroll
        for (int ni = 0; ni < 4; ++ni)
            sK[row * LDSD + ni * 16 + col] = oacc[ni][e];
    }
    __syncthreads();
    {
        const int c4 = t & 15;
        const int r0 = t >> 4;
        for (int i = 0; i < 8; ++i) {
            int r = r0 + 8 * i;
            float4 ov = *(const float4*)&sK[r * LDSD + c4 * 4];
            ((float4*)(obuf + (bhS + sQI[r]) * D))[c4] = ov;
        }
    }
}

// ---------------------------------------------------------------------------
// Stage 4: softmax over hashes and weighted sum.  grid = bs*S, 64 threads.
// ---------------------------------------------------------------------------
__global__ __launch_bounds__(64)
void combine_kernel(const float* __restrict__ obuf,
                    const float* __restrict__ logits,
                    float* __restrict__ out,
                    int bs, int S)
{
    const int blk = blockIdx.x;
    const int b   = blk / S;
    const int pos = blk % S;
    const int d   = threadIdx.x;

    float l[NHASH];
    float mx = -3.4e38f;
    #pragma unroll
    for (int h = 0; h < NHASH; ++h) {
        l[h] = logits[((size_t)(b * NHASH + h)) * S + pos];
        mx = fmaxf(mx, l[h]);
    }
    float s = 0.f;
    #pragma unroll
    for (int h = 0; h < NHASH; ++h) { l[h] = __expf(l[h] - mx); s += l[h]; }
    const float inv = 1.0f / s;
    float v = 0.f;
    #pragma unroll
    for (int h = 0; h < NHASH; ++h)
        v += obuf[(((size_t)(b * NHASH + h)) * S + pos) * D + d] * l[h];
    out[((size_t)b * S + pos) * D + d] = v * inv;
}

// ---------------------------------------------------------------------------
extern "C" void kernel_launch(void* const* d_in, const int* in_sizes, int n_in,
                              void* d_out, int out_size, void* d_ws, size_t ws_size,
                              hipStream_t stream) {
    const float* query   = (const float*)d_in[0];
    const float* keym    = (const float*)d_in[1];
    const float* value   = (const float*)d_in[2];
    const float* alpha_q = (const float*)d_in[3];
    const float* beta_q  = (const float*)d_in[4];
    const float* w_q     = (const float*)d_in[5];
    const float* alpha_k = (const float*)d_in[6];
    const float* beta_k  = (const float*)d_in[7];
    const float* w_k     = (const float*)d_in[8];
    float* out = (float*)d_out;

    const int NB = in_sizes[5];          // n_buckets (== 64)
    const int S  = NB * BKT;             // seqlen (== 4096)
    const int bs = in_sizes[0] / (S * D);

    const size_t nBH = (size_t)bs * NHASH * S;
    char* ws = (char*)d_ws;
    int*   qb  = (int*)ws;   ws += nBH * sizeof(int);
    int*   kb  = (int*)ws;   ws += nBH * sizeof(int);
    int*   qst = (int*)ws;   ws += nBH * sizeof(int);
    int*   kst = (int*)ws;   ws += nBH * sizeof(int);
    float* lg  = (float*)ws; ws += nBH * sizeof(float);
    float* ob  = (float*)ws;                 // nBH * D floats

    alsh_hash_kernel<<<2 * bs * (S >> 6), 512, 0, stream>>>(
        query, keym, alpha_q, beta_q, w_q, alpha_k, beta_k, w_k, qb, kb, bs, S);

    stable_bucket_sort_kernel<<<2 * bs * NHASH, 256, 0, stream>>>(
        qb, kb, qst, kst, bs, S);

    lsh_attn_chunk_kernel<<<bs * NHASH * (S >> 6), 128, 0, stream>>>(
        query, keym, value, qst, kst, lg, ob, bs, S);

    combine_kernel<<<bs * S, 64, 0, stream>>>(ob, lg, out, bs, S);
}